// GCN_17386027614906
// MI455X (gfx1250) — compile-verified
//
#include <hip/hip_runtime.h>
#include <hip/hip_bf16.h>
#include <math.h>

// ---------------------------------------------------------------------------
// GCN 3-layer inference for MI455X (gfx1250, wave32).
//  - GEMMs via V_WMMA_F32_16X16X4_F32 (full fp32 precision, K=128 -> 32 wmma/tile)
//  - W staged into LDS with GLOBAL_LOAD_ASYNC_TO_LDS_B128 (ASYNCcnt path)
//  - Edge scatter via native global_atomic_add_f32 (inline asm, scope:SCOPE_DEV)
//    -> guaranteed L2 atomic-unit RMW, no CAS-loop fallback. All feature maps
//    (25.6MB) are L2-resident (192MB), so gather+atomic traffic stays on-chip.
//  - Fused self-loop + bias + BN + ReLU; wave32 shuffle log-softmax.
// ---------------------------------------------------------------------------

typedef __attribute__((ext_vector_type(2))) float v2f;
typedef __attribute__((ext_vector_type(8))) float v8f;

// Native fp32 atomic add at device scope (L2 atomic units). No return form.
__device__ __forceinline__ void atomic_add_f32(float* p, float v) {
    asm volatile("global_atomic_add_f32 %0, %1, off scope:SCOPE_DEV"
                 :: "v"(p), "v"(v)
                 : "memory");
}

// ---------------------------------------------------------------------------
// Utility fills
// ---------------------------------------------------------------------------
__global__ __launch_bounds__(256) void fill_f32(float* __restrict__ p, float val, int n) {
    int i = blockIdx.x * 256 + threadIdx.x;
    if (i < n) p[i] = val;
}

__global__ __launch_bounds__(256) void deg_count(const int* __restrict__ dst,
                                                 float* __restrict__ deg, int nE) {
    int e = blockIdx.x * 256 + threadIdx.x;
    if (e < nE) atomic_add_f32(deg + dst[e], 1.0f);
}

__global__ __launch_bounds__(256) void deg_finish(const float* __restrict__ deg,
                                                  float* __restrict__ dis,
                                                  float* __restrict__ invd, int n) {
    int i = blockIdx.x * 256 + threadIdx.x;
    if (i < n) {
        float dg = deg[i];
        dis[i]  = rsqrtf(dg);
        invd[i] = 1.0f / dg;
    }
}

// ---------------------------------------------------------------------------
// GEMM: H[nrows x NPAD] = X[nrows x 128] @ W[128 x ntrue]  (cols >= ntrue zero)
// One wave computes a 16-row slab across all NPAD output columns.
// A-frag layout: lanes 0-15 -> (M=lane, K=k..k+1); lanes 16-31 -> (M=lane-16, K=k+2..k+3)
// B-frag layout mirrored (rows K=k.. striped same way), read from LDS copy of W.
// C/D layout: VGPR r, lanes 0-15 -> M=r, lanes 16-31 -> M=8+r, N=lane&15 (+16*t).
// ---------------------------------------------------------------------------
template<int NPAD>
__global__ __launch_bounds__(256)
void gemm_wmma_f32(const float* __restrict__ X, const float* __restrict__ W,
                   float* __restrict__ H, int nrows, int ntrue) {
    __shared__ __align__(16) float lw[128 * NPAD];
    const int tid = threadIdx.x;

    if (ntrue == NPAD) {
        // CDNA5 async DMA straight into LDS: 16B per lane per issue.
        // Low 32 bits of the flat LDS-aperture address == LDS byte offset (ISA 10.2).
        const unsigned ldsbase = (unsigned)(size_t)(void*)lw;
        for (int c = tid; c < (128 * NPAD) / 4; c += 256) {
            const float* gp = W + (size_t)c * 4;
            const unsigned loff = ldsbase + (unsigned)c * 16u;
            asm volatile("global_load_async_to_lds_b128 %0, %1, off"
                         :: "v"(loff), "v"(gp) : "memory");
        }
        asm volatile("s_wait_asynccnt 0" ::: "memory");
    } else {
        for (int idx = tid; idx < 128 * NPAD; idx += 256) {
            int kk = idx / NPAD, nn = idx % NPAD;
            lw[idx] = (nn < ntrue) ? W[kk * ntrue + nn] : 0.0f;
        }
    }
    __syncthreads();

    const int wave = tid >> 5;
    const int lane = tid & 31;
    const int row0 = (blockIdx.x * 8 + wave) * 16;
    if (row0 >= nrows) return;            // wave-uniform: EXEC stays all-ones

    const int half = lane >> 4;           // 0 | 1 (branchless)
    const int l16  = lane & 15;
    const float* xrow = X + (size_t)(row0 + l16) * 128;

    constexpr int NT = NPAD / 16;
    v8f acc[NT];
#pragma unroll
    for (int t = 0; t < NT; ++t)
#pragma unroll
        for (int i = 0; i < 8; ++i) acc[t][i] = 0.0f;

    for (int k = 0; k < 128; k += 4) {
        const int ka = k + half * 2;
        v2f a;
        a.x = xrow[ka];
        a.y = xrow[ka + 1];
#pragma unroll
        for (int t = 0; t < NT; ++t) {
            const int n = t * 16 + l16;
            v2f b;
            b.x = lw[ka * NPAD + n];
            b.y = lw[(ka + 1) * NPAD + n];
            acc[t] = __builtin_amdgcn_wmma_f32_16x16x4_f32(
                false, a, false, b, (short)0, acc[t], false, false);
        }
    }

#pragma unroll
    for (int t = 0; t < NT; ++t)
#pragma unroll
        for (int r = 0; r < 8; ++r)
            H[(size_t)(row0 + half * 8 + r) * NPAD + t * 16 + l16] = acc[t][r];
}

// ---------------------------------------------------------------------------
// Edge aggregation, D=128: one wave per edge, lane handles 4 floats (float4).
// agg[dst] += h[src] * dis[src]*dis[dst]  -> 4 global_atomic_add_f32 per lane.
// ---------------------------------------------------------------------------
__global__ __launch_bounds__(256)
void edge_agg_d128(const int* __restrict__ src, const int* __restrict__ dst,
                   const float* __restrict__ dis, const float* __restrict__ h,
                   float* __restrict__ agg, int nE) {
    const int lane = threadIdx.x & 31;
    const int e = (blockIdx.x * 256 + threadIdx.x) >> 5;
    if (e >= nE) return;
    const int s = src[e], d = dst[e];
    const float norm = dis[s] * dis[d];
    const float4 hv = *((const float4*)(h + (size_t)s * 128) + lane);
    float* ap = agg + (size_t)d * 128 + lane * 4;
    atomic_add_f32(ap + 0, hv.x * norm);
    atomic_add_f32(ap + 1, hv.y * norm);
    atomic_add_f32(ap + 2, hv.z * norm);
    atomic_add_f32(ap + 3, hv.w * norm);
}

// Edge aggregation, padded D=48: one wave per edge; lane covers col lane and
// (for lanes 0-15) col lane+32.
__global__ __launch_bounds__(256)
void edge_agg_d48(const int* __restrict__ src, const int* __restrict__ dst,
                  const float* __restrict__ dis, const float* __restrict__ h,
                  float* __restrict__ agg, int nE) {
    const int lane = threadIdx.x & 31;
    const int e = (blockIdx.x * 256 + threadIdx.x) >> 5;
    if (e >= nE) return;
    const int s = src[e], d = dst[e];
    const float norm = dis[s] * dis[d];
    const float* hp = h + (size_t)s * 48;
    float* ap = agg + (size_t)d * 48;
    atomic_add_f32(ap + lane, hp[lane] * norm);
    if (lane < 16) atomic_add_f32(ap + 32 + lane, hp[32 + lane] * norm);
}

// ---------------------------------------------------------------------------
// agg = relu( bn( agg + h * invdeg + bias ) )  — in place on agg, D=128.
// ---------------------------------------------------------------------------
__global__ __launch_bounds__(256)
void finalize_bn_relu(float* __restrict__ agg, const float* __restrict__ h,
                      const float* __restrict__ invdeg, const float* __restrict__ b,
                      const float* __restrict__ g, const float* __restrict__ be,
                      const float* __restrict__ m, const float* __restrict__ v, int nN) {
    const int idx = blockIdx.x * 256 + threadIdx.x;
    if (idx >= nN * 128) return;
    const int i = idx >> 7, j = idx & 127;
    float val = agg[idx] + h[idx] * invdeg[i] + b[j];
    val = (val - m[j]) * rsqrtf(v[j] + 1e-5f) * g[j] + be[j];
    agg[idx] = fmaxf(val, 0.0f);
}

// ---------------------------------------------------------------------------
// Layer-3 finalize + log_softmax over 40 classes. One wave32 per row.
// Lane handles col=lane (always <40) and, for lanes 0-7, col=lane+32.
// ---------------------------------------------------------------------------
__global__ __launch_bounds__(256)
void final3_logsoftmax(const float* __restrict__ agg, const float* __restrict__ h,
                       const float* __restrict__ invdeg, const float* __restrict__ b,
                       float* __restrict__ out, int nN) {
    const int lane = threadIdx.x & 31;
    const int row = (blockIdx.x * 256 + threadIdx.x) >> 5;
    if (row >= nN) return;
    const float* ar = agg + (size_t)row * 48;
    const float* hr = h + (size_t)row * 48;
    const float inv = invdeg[row];

    const float z0 = ar[lane] + hr[lane] * inv + b[lane];
    float z1 = 0.0f;
    if (lane < 8) z1 = ar[32 + lane] + hr[32 + lane] * inv + b[32 + lane];

    float mx = z0;
    if (lane < 8) mx = fmaxf(mx, z1);
    for (int off = 16; off > 0; off >>= 1) mx = fmaxf(mx, __shfl_xor(mx, off, 32));

    float se = __expf(z0 - mx);
    if (lane < 8) se += __expf(z1 - mx);
    for (int off = 16; off > 0; off >>= 1) se += __shfl_xor(se, off, 32);

    const float lse = __logf(se) + mx;
    float* orow = out + (size_t)row * 40;
    orow[lane] = z0 - lse;
    if (lane < 8) orow[32 + lane] = z1 - lse;
}

// ---------------------------------------------------------------------------
extern "C" void kernel_launch(void* const* d_in, const int* in_sizes, int n_in,
                              void* d_out, int out_size, void* d_ws, size_t ws_size,
                              hipStream_t stream) {
    const float* x   = (const float*)d_in[0];
    const int*   ei  = (const int*)d_in[1];
    const float* W1  = (const float*)d_in[2];
    const float* b1  = (const float*)d_in[3];
    const float* W2  = (const float*)d_in[4];
    const float* b2  = (const float*)d_in[5];
    const float* W3  = (const float*)d_in[6];
    const float* b3  = (const float*)d_in[7];
    const float* g1  = (const float*)d_in[8];
    const float* be1 = (const float*)d_in[9];
    const float* m1  = (const float*)d_in[10];
    const float* v1  = (const float*)d_in[11];
    const float* g2  = (const float*)d_in[12];
    const float* be2 = (const float*)d_in[13];
    const float* m2  = (const float*)d_in[14];
    const float* v2  = (const float*)d_in[15];
    float* out = (float*)d_out;

    const int Nn = in_sizes[0] / 128;   // 50000
    const int Ee = in_sizes[1] / 2;     // 800000
    const int* srcp = ei;
    const int* dstp = ei + Ee;

    // workspace layout (floats): deg[N] dis[N] invd[N] bufA[N*128] bufB[N*128]
    float* wsf  = (float*)d_ws;
    float* deg  = wsf;
    float* dis  = deg + Nn;
    float* invd = dis + Nn;
    float* bufA = invd + Nn;
    float* bufB = bufA + (size_t)Nn * 128;

    const int TB = 256;
    const int nBlkN   = (Nn + TB - 1) / TB;
    const int nBlkE   = (Ee + TB - 1) / TB;
    const int nBlkNF  = (Nn * 128 + TB - 1) / TB;
    const int gemmBlk = ((Nn + 15) / 16 + 7) / 8;   // 16 rows/wave, 8 waves/block
    const int edgeBlk = (Ee + 7) / 8;               // 1 wave/edge, 8 waves/block
    const int rowBlk  = (Nn + 7) / 8;               // 1 wave/row

    // degrees (shared by all layers)
    fill_f32  <<<nBlkN, TB, 0, stream>>>(deg, 1.0f, Nn);   // self loop
    deg_count <<<nBlkE, TB, 0, stream>>>(dstp, deg, Ee);
    deg_finish<<<nBlkN, TB, 0, stream>>>(deg, dis, invd, Nn);

    // ---- layer 1 ----
    gemm_wmma_f32<128><<<gemmBlk, TB, 0, stream>>>(x, W1, bufA, Nn, 128);
    fill_f32         <<<nBlkNF, TB, 0, stream>>>(bufB, 0.0f, Nn * 128);
    edge_agg_d128    <<<edgeBlk, TB, 0, stream>>>(srcp, dstp, dis, bufA, bufB, Ee);
    finalize_bn_relu <<<nBlkNF, TB, 0, stream>>>(bufB, bufA, invd, b1, g1, be1, m1, v1, Nn);

    // ---- layer 2 ----  (bufB = x2 consumed by GEMM before it is re-zeroed)
    gemm_wmma_f32<128><<<gemmBlk, TB, 0, stream>>>(bufB, W2, bufA, Nn, 128);
    fill_f32         <<<nBlkNF, TB, 0, stream>>>(bufB, 0.0f, Nn * 128);
    edge_agg_d128    <<<edgeBlk, TB, 0, stream>>>(srcp, dstp, dis, bufA, bufB, Ee);
    finalize_bn_relu <<<nBlkNF, TB, 0, stream>>>(bufB, bufA, invd, b2, g2, be2, m2, v2, Nn);

    // ---- layer 3 ----  (40 outputs padded to 48)
    gemm_wmma_f32<48><<<gemmBlk, TB, 0, stream>>>(bufB, W3, bufA, Nn, 40);
    fill_f32         <<<(Nn * 48 + TB - 1) / TB, TB, 0, stream>>>(bufB, 0.0f, Nn * 48);
    edge_agg_d48     <<<edgeBlk, TB, 0, stream>>>(srcp, dstp, dis, bufA, bufB, Ee);
    final3_logsoftmax<<<rowBlk, TB, 0, stream>>>(bufB, bufA, invd, b3, out, Nn);
}